// Linear_diagonal_weight_69836168232969
// MI455X (gfx1250) — compile-verified
//
#include <hip/hip_runtime.h>

// out[i, j] = input[i, j] * weight[j, j] + bias[j]
// B = 8192 rows, N = 4096 cols, all fp32.
// Memory-bound: ~256 MiB of streaming traffic, ~11.5 us floor at 23.3 TB/s.

#define N_COLS     4096
#define B_ROWS     8192
#define COLS4      (N_COLS / 4)        // 1024 float4 column slots
#define ROWS_PER_T 8                   // rows streamed per thread
#define TPB        256                 // 8 wave32 per block

typedef __attribute__((ext_vector_type(4))) float v4f;

// Stage 1: pack diagonal of weight into contiguous scratch (16 KiB, L2-resident).
__global__ __launch_bounds__(256) void gather_diag_kernel(
    const float* __restrict__ weight, float* __restrict__ dw) {
  int j = blockIdx.x * blockDim.x + threadIdx.x;
  if (j < N_COLS) {
    dw[j] = weight[(size_t)j * (size_t)(N_COLS + 1)];
  }
}

// Stage 2: streaming scale + bias. Non-temporal 128-bit loads/stores on the
// 128 MiB input/output (touched exactly once; together they exceed the
// 192 MB L2, so NT keeps diag/bias tables hot instead).
__global__ __launch_bounds__(TPB) void scale_bias_kernel(
    const v4f* __restrict__ in, const v4f* __restrict__ dw4,
    const v4f* __restrict__ bias4, v4f* __restrict__ out) {
  const int col4 = blockIdx.x * TPB + threadIdx.x;        // 0 .. COLS4-1
  const int row0 = blockIdx.y * ROWS_PER_T;               // 0 .. B_ROWS-8

  // Per-thread constants: loaded once, regular (temporal) path -> WGP$/L2 hit.
  const v4f w4 = dw4[col4];
  const v4f b4 = bias4[col4];

  size_t idx = (size_t)row0 * COLS4 + (size_t)col4;
#pragma unroll
  for (int r = 0; r < ROWS_PER_T; ++r) {
    v4f v = __builtin_nontemporal_load(&in[idx]);
    v4f o;
    o.x = fmaf(v.x, w4.x, b4.x);
    o.y = fmaf(v.y, w4.y, b4.y);
    o.z = fmaf(v.z, w4.z, b4.z);
    o.w = fmaf(v.w, w4.w, b4.w);
    __builtin_nontemporal_store(o, &out[idx]);
    idx += COLS4;
  }
}

extern "C" void kernel_launch(void* const* d_in, const int* in_sizes, int n_in,
                              void* d_out, int out_size, void* d_ws, size_t ws_size,
                              hipStream_t stream) {
  const float* input  = (const float*)d_in[0];   // [B_ROWS, N_COLS]
  const float* weight = (const float*)d_in[1];   // [N_COLS, N_COLS]
  const float* bias   = (const float*)d_in[2];   // [N_COLS]
  float* out = (float*)d_out;                    // [B_ROWS, N_COLS]
  float* dw  = (float*)d_ws;                     // 4096 floats = 16 KiB scratch

  // Stage 1: diag(weight) -> contiguous scratch.
  gather_diag_kernel<<<N_COLS / 256, 256, 0, stream>>>(weight, dw);

  // Stage 2: streaming elementwise FMA.
  dim3 grid(COLS4 / TPB, B_ROWS / ROWS_PER_T);   // (4, 1024)
  scale_bias_kernel<<<grid, TPB, 0, stream>>>(
      (const v4f*)input, (const v4f*)dw, (const v4f*)bias, (v4f*)out);
}